// CA1AttentionGate_21809843929412
// MI455X (gfx1250) — compile-verified
//
#include <hip/hip_runtime.h>
#include <stdint.h>

// ---------------------------------------------------------------------------
// CA1AttentionGate for MI455X (gfx1250, wave32).
// Bandwidth-bound: ~64MB reads + 256MB broadcast write => ~14us roofline.
//  - row = g @ key^T via V_WMMA_F32_16X16X4_F32 (fp32-native WMMA)
//  - key tiles staged with GLOBAL_LOAD_ASYNC_TO_LDS_B128 (ASYNCcnt path)
//  - output broadcast: register-cached rows + nontemporal b128 stores
// ---------------------------------------------------------------------------

constexpr int F_DIM  = 128;
constexpr int H_DIM  = 1024;
constexpr int SEQ    = 8192;
constexpr int K_ANCH = 2;
constexpr int NSEG   = 16;          // row-segments for deterministic qmean

// workspace layout (float offsets)
constexpr size_t WS_PARTIAL = 0;                                 // NSEG * H_DIM
constexpr size_t WS_WEIGHT  = WS_PARTIAL + (size_t)NSEG * H_DIM; // 512
constexpr size_t WS_G       = WS_WEIGHT + 512;                   // K_ANCH * H_DIM
constexpr size_t WS_MEANROW = WS_G + (size_t)K_ANCH * H_DIM;     // SEQ

typedef __attribute__((ext_vector_type(2))) float v2f;
typedef __attribute__((ext_vector_type(4))) float v4f;
typedef __attribute__((ext_vector_type(8))) float v8f;

// ---------------------------------------------------------------------------
// 1) Partial column sums of query over S (deterministic, no atomics).
//    grid = (H_DIM/256, NSEG), block = 256
// ---------------------------------------------------------------------------
__global__ void __launch_bounds__(256) ca1_qsum_kernel(
    const float* __restrict__ query, float* __restrict__ partial)
{
    const int h   = blockIdx.x * 256 + threadIdx.x;
    const int seg = blockIdx.y;
    const int s0  = seg * (SEQ / NSEG);
    const float* p = query + (size_t)s0 * H_DIM + h;
    float acc = 0.f;
    for (int s = 0; s < SEQ / NSEG; ++s)
        acc += p[(size_t)s * H_DIM];
    partial[seg * H_DIM + h] = acc;
}

// ---------------------------------------------------------------------------
// 2) Everything tiny: qmean finish, temporal MLP, attention gate, g = mem@Wg.
//    grid = 1, block = 128
// ---------------------------------------------------------------------------
__global__ void __launch_bounds__(128) ca1_prep_kernel(
    const float* __restrict__ partial, const float* __restrict__ dg,
    const float* __restrict__ ts,  const float* __restrict__ Wt1,
    const float* __restrict__ bt1, const float* __restrict__ Wt2,
    const float* __restrict__ bt2, const float* __restrict__ Wa1,
    const float* __restrict__ ba1, const float* __restrict__ Wa2,
    const float* __restrict__ ba2, const float* __restrict__ Wg,
    const float* __restrict__ bg,
    float* __restrict__ g_out, float* __restrict__ w_out)
{
    __shared__ float sh_q[H_DIM];
    __shared__ float sh_mem[K_ANCH][F_DIM];
    __shared__ float sh_hid[32];
    __shared__ float sh_red[F_DIM];
    const int tid = threadIdx.x;

    // qmean
    for (int h = tid; h < H_DIM; h += 128) {
        float acc = 0.f;
        for (int s = 0; s < NSEG; ++s) acc += partial[s * H_DIM + h];
        sh_q[h] = acc * (1.0f / (float)SEQ);
    }
    __syncthreads();

    // temporal MLP + mem = dg + temporal
    for (int k = 0; k < K_ANCH; ++k) {
        if (tid < 32)
            sh_hid[tid] = fmaxf(ts[k] * Wt1[tid] + bt1[tid], 0.f);
        __syncthreads();
        float acc = bt2[tid];
        for (int j = 0; j < 32; ++j)
            acc += sh_hid[j] * Wt2[j * F_DIM + tid];
        sh_mem[k][tid] = dg[k * F_DIM + tid] + acc;
        __syncthreads();
    }

    // gate: weight[k] = sigmoid(tanh([mem;qmean] @ Wa1 + ba1) @ Wa2 + ba2)
    for (int k = 0; k < K_ANCH; ++k) {
        float pre = ba1[tid];
        for (int d = 0; d < F_DIM; ++d)
            pre += sh_mem[k][d] * Wa1[d * F_DIM + tid];
        for (int d = 0; d < H_DIM; ++d)
            pre += sh_q[d] * Wa1[(F_DIM + d) * F_DIM + tid];
        sh_red[tid] = tanhf(pre) * Wa2[tid];
        __syncthreads();
        for (int off = 64; off > 0; off >>= 1) {
            if (tid < off) sh_red[tid] += sh_red[tid + off];
            __syncthreads();
        }
        if (tid == 0) {
            const float score = sh_red[0] + ba2[0];
            w_out[k] = 1.0f / (1.0f + expf(-score));
        }
        __syncthreads();
    }

    // g = mem @ Wg + bg   (2 x 1024)
    for (int k = 0; k < K_ANCH; ++k)
        for (int h = tid; h < H_DIM; h += 128) {
            float acc = bg[h];
            for (int f = 0; f < F_DIM; ++f)
                acc += sh_mem[k][f] * Wg[f * H_DIM + h];
            g_out[k * H_DIM + h] = acc;
        }
}

// ---------------------------------------------------------------------------
// 3) row[k,s] = dot(g[k], key[s]); mean_row[s] = (1/K) sum_k w[k]*row[k,s]
//    WMMA f32 16x16x4: M = anchors (padded, unused rows ignored),
//    N = 16 seq positions per wave, K = H in steps of 4.
//    Key tiles staged global->LDS with GLOBAL_LOAD_ASYNC_TO_LDS_B128.
//    grid = SEQ/128, block = 256 (8 waves, one 16-seq tile each)
// ---------------------------------------------------------------------------
__global__ void __launch_bounds__(256) ca1_row_wmma_kernel(
    const float* __restrict__ key, const float* __restrict__ g,
    const float* __restrict__ w,   float* __restrict__ mean_row)
{
    __shared__ float gsh[K_ANCH * H_DIM];
    __shared__ __align__(16) float ktile[8][16][68];   // pitch 68: conflict-free b64 reads
    const int tid = threadIdx.x;

    for (int i = tid; i < K_ANCH * H_DIM; i += 256)
        gsh[i] = g[i];
    __syncthreads();

    const int wave = tid >> 5;
    const int lane = tid & 31;
    const int s0   = blockIdx.x * 128 + wave * 16;
    const int m    = lane & 15;             // A-row (M) / B-col (N) index
    const int koff = (lane >> 4) << 1;      // K pair: lanes 0-15 -> K0/K1, 16-31 -> K2/K3

    const int rbase  = lane >> 4;           // staging: row base within pair
    const int cg     = lane & 15;           // staging: 16B column group

    v8f acc = {};
    for (int h0 = 0; h0 < H_DIM; h0 += 64) {
        // Previous iteration's B-fragment ds_loads must drain before the async
        // writes land (DScnt and ASYNCcnt are mutually unordered).
        asm volatile("s_wait_dscnt 0x0" ::: "memory");

        // Stage 16 x 64 key tile straight into LDS: 8 in-flight async b128
        // transfers per lane group, no VGPR round-trip.
        #pragma unroll
        for (int it = 0; it < 8; ++it) {
            const int r = rbase + it * 2;
            const float* src = key + (size_t)(s0 + r) * H_DIM + h0 + cg * 4;
            const unsigned dst = (unsigned)(uintptr_t)&ktile[wave][r][cg * 4];
            asm volatile("global_load_async_to_lds_b128 %0, %1, off"
                         :: "v"(dst), "v"(src) : "memory");
        }
        asm volatile("s_wait_asynccnt 0x0" ::: "memory");

        const float* grow = &gsh[(m & (K_ANCH - 1)) * H_DIM + h0];
        #pragma unroll
        for (int c = 0; c < 16; ++c) {
            v2f a, b;
            // A 16x4 layout: v0 = K0 (lanes 0-15) / K2 (16-31); v1 = K1 / K3
            a.x = grow[c * 4 + koff];
            a.y = grow[c * 4 + koff + 1];
            // B 4x16 layout mirrors A: lane -> column n, koff -> K rows
            b.x = ktile[wave][m][c * 4 + koff];
            b.y = ktile[wave][m][c * 4 + koff + 1];
            acc = __builtin_amdgcn_wmma_f32_16x16x4_f32(
                false, a, false, b, (short)0, acc, false, false);
        }
    }

    // D layout: lane n (<16), VGPR r holds row[M=r, N=n]
    if (lane < 16) {
        const float w0 = w[0], w1 = w[1];
        mean_row[s0 + lane] = 0.5f * (w0 * acc[0] + w1 * acc[1]);
    }
}

// ---------------------------------------------------------------------------
// 4) Broadcast mean_row to (1,1,S,S): 256 MB of pure stores (the real cost).
//    Cache the whole row in registers, write 8 rows per block.
//    grid = SEQ/8, block = 256; nontemporal 128-bit stores.
// ---------------------------------------------------------------------------
constexpr int ROWS_PER_BLK = 8;

__global__ void __launch_bounds__(256) ca1_bcast_kernel(
    const float* __restrict__ mean_row, float* __restrict__ out)
{
    const int tid = threadIdx.x;
    v4f vals[8];
    #pragma unroll
    for (int c = 0; c < 8; ++c)
        vals[c] = *(const v4f*)(mean_row + c * 1024 + tid * 4);

    const size_t r0 = (size_t)blockIdx.x * ROWS_PER_BLK;
    #pragma unroll
    for (int r = 0; r < ROWS_PER_BLK; ++r) {
        float* dst = out + (r0 + r) * SEQ;
        #pragma unroll
        for (int c = 0; c < 8; ++c)
            __builtin_nontemporal_store(vals[c], (v4f*)(dst + c * 1024 + tid * 4));
    }
}

// ---------------------------------------------------------------------------
extern "C" void kernel_launch(void* const* d_in, const int* in_sizes, int n_in,
                              void* d_out, int out_size, void* d_ws, size_t ws_size,
                              hipStream_t stream)
{
    const float* query = (const float*)d_in[0];
    const float* key   = (const float*)d_in[1];
    const float* dg    = (const float*)d_in[2];
    const float* ts    = (const float*)d_in[3];
    const float* Wt1   = (const float*)d_in[4];
    const float* bt1   = (const float*)d_in[5];
    const float* Wt2   = (const float*)d_in[6];
    const float* bt2   = (const float*)d_in[7];
    const float* Wa1   = (const float*)d_in[8];
    const float* ba1   = (const float*)d_in[9];
    const float* Wa2   = (const float*)d_in[10];
    const float* ba2   = (const float*)d_in[11];
    const float* Wg    = (const float*)d_in[12];
    const float* bg    = (const float*)d_in[13];

    float* ws      = (float*)d_ws;
    float* partial = ws + WS_PARTIAL;
    float* weight  = ws + WS_WEIGHT;
    float* gbuf    = ws + WS_G;
    float* meanrow = ws + WS_MEANROW;
    float* out     = (float*)d_out;

    ca1_qsum_kernel<<<dim3(H_DIM / 256, NSEG), 256, 0, stream>>>(query, partial);
    ca1_prep_kernel<<<1, 128, 0, stream>>>(partial, dg, ts, Wt1, bt1, Wt2, bt2,
                                           Wa1, ba1, Wa2, ba2, Wg, bg,
                                           gbuf, weight);
    ca1_row_wmma_kernel<<<SEQ / 128, 256, 0, stream>>>(key, gbuf, weight, meanrow);
    ca1_bcast_kernel<<<SEQ / ROWS_PER_BLK, 256, 0, stream>>>(meanrow, out);
}